// EdgeTypeClassifier_76424648065478
// MI455X (gfx1250) — compile-verified
//
#include <hip/hip_runtime.h>

typedef float v2f __attribute__((ext_vector_type(2)));
typedef float v8f __attribute__((ext_vector_type(8)));

#define HID  128
#define NTYPE 8
#define ROWPAD 132            // 128 + 4 floats pad -> conflict-free strided LDS reads
#define WAVES_PER_BLOCK 4
#define TILE_M 16

#if __has_builtin(__builtin_amdgcn_sched_barrier)
#define SCHED_FENCE() __builtin_amdgcn_sched_barrier(0)
#else
#define SCHED_FENCE() asm volatile("" ::: "memory")
#endif

__global__ __launch_bounds__(WAVES_PER_BLOCK * 32)
void edge_type_wmma_kernel(const float* __restrict__ X,
                           const long long* __restrict__ EI,
                           const float* __restrict__ W,
                           const float* __restrict__ Bv,
                           float* __restrict__ out,
                           int E, int numTiles, int totalWaves) {
    __shared__ float hbuf[WAVES_PER_BLOCK][TILE_M][ROWPAD];
    __shared__ float wlds[16][ROWPAD];   // W^T staging, rows 8..15 zero

    const int tid    = threadIdx.x;
    const int lane   = tid & 31;
    const int wave   = tid >> 5;
    const int waveId = blockIdx.x * WAVES_PER_BLOCK + wave;

    const int n    = lane & 15;   // N column (edge-type id when < 8)
    const int hi   = lane >> 4;   // selects K-pair {0,1} vs {2,3}
    const int koff = hi * 2;

    // ---- Stage W into LDS once per block (rows >= NTYPE zero-filled) ----
    for (int idx = tid; idx < 16 * HID; idx += WAVES_PER_BLOCK * 32) {
        const int r = idx >> 7;       // / HID
        const int c = idx & (HID - 1);
        wlds[r][c] = (r < NTYPE) ? W[r * HID + c] : 0.f;
    }
    const float bias = (n < NTYPE) ? Bv[n] : 0.f;
    __syncthreads();

    float*       myrow0 = &hbuf[wave][0][0];
    const float* wrow   = &wlds[n][0];

    for (int t = waveId; t < numTiles; t += totalWaves) {
        const int base = t * TILE_M;

        // lanes 0..15 hold (src,dst) for edges base..base+15 (branch-free clamp);
        // indices < N_NODES, so the low 32 bits address correctly.
        int eLane = base + n;
        eLane = (eLane < E) ? eLane : (E - 1);
        const int sLane = (int)EI[eLane];
        const int dLane = (int)EI[(long long)E + eLane];

        // ---- Phase 1: h = relu(x[src] + x[dst]) -> LDS ----
        // Groups of 8 edges: 16 SADDR-form b128 gathers in flight before any
        // consumption; sched_barrier pins the issue order.
#pragma unroll
        for (int g = 0; g < TILE_M; g += 8) {
            float4 av[8], bv[8];
#pragma unroll
            for (int e = 0; e < 8; ++e) {
                const unsigned s = (unsigned)__builtin_amdgcn_readlane(sLane, g + e);
                const unsigned d = (unsigned)__builtin_amdgcn_readlane(dLane, g + e);
                av[e] = ((const float4*)(X + (size_t)s * HID))[lane];
                bv[e] = ((const float4*)(X + (size_t)d * HID))[lane];
            }
            SCHED_FENCE();   // all 16 loads issued before first consume
#pragma unroll
            for (int e = 0; e < 8; ++e) {
                float4 h;
                h.x = fmaxf(av[e].x + bv[e].x, 0.f);
                h.y = fmaxf(av[e].y + bv[e].y, 0.f);
                h.z = fmaxf(av[e].z + bv[e].z, 0.f);
                h.w = fmaxf(av[e].w + bv[e].w, 0.f);
                *(float4*)(myrow0 + (g + e) * ROWPAD + lane * 4) = h;
            }
        }

        // LDS visibility within the wave before the strided fragment reads
        asm volatile("s_wait_dscnt 0" ::: "memory");

        // ---- Phase 2: K=128 via 32 x v_wmma_f32_16x16x4_f32, dual accumulators,
        //      A and B fragments both from padded LDS (conflict-free) ----
        v8f acc0 = {};
        v8f acc1 = {};
#pragma unroll
        for (int ks = 0; ks < HID / 4; ks += 2) {
            v2f a0 = *(const v2f*)(myrow0 + n * ROWPAD + ks * 4 + koff);
            v2f a1 = *(const v2f*)(myrow0 + n * ROWPAD + (ks + 1) * 4 + koff);
            v2f b0 = *(const v2f*)(wrow + ks * 4 + koff);
            v2f b1 = *(const v2f*)(wrow + (ks + 1) * 4 + koff);
            acc0 = __builtin_amdgcn_wmma_f32_16x16x4_f32(
                false, a0, false, b0, (short)0, acc0, false, false);
            acc1 = __builtin_amdgcn_wmma_f32_16x16x4_f32(
                false, a1, false, b1, (short)0, acc1, false, false);
        }

        // ---- Store: acc[j] = D[j + 8*hi][n] -> logits[base + j + 8*hi][n] ----
        if (n < NTYPE) {
            if (base + TILE_M <= E) {
#pragma unroll
                for (int j = 0; j < 8; ++j) {
                    const int row = base + j + 8 * hi;
                    out[row * NTYPE + n] = acc0[j] + acc1[j] + bias;
                }
            } else {
#pragma unroll
                for (int j = 0; j < 8; ++j) {
                    const int row = base + j + 8 * hi;
                    if (row < E)
                        out[row * NTYPE + n] = acc0[j] + acc1[j] + bias;
                }
            }
        }
    }
}

extern "C" void kernel_launch(void* const* d_in, const int* in_sizes, int n_in,
                              void* d_out, int out_size, void* d_ws, size_t ws_size,
                              hipStream_t stream) {
    const float*     X   = (const float*)d_in[0];
    const long long* EI  = (const long long*)d_in[1];  // int64 edge_index [2, E]
    const float*     W   = (const float*)d_in[2];      // [8, 128]
    const float*     Bv  = (const float*)d_in[3];      // [8]
    float*           out = (float*)d_out;              // [E, 8]

    const int E        = in_sizes[1] / 2;
    const int numTiles = (E + TILE_M - 1) / TILE_M;

    int blocks = (numTiles + WAVES_PER_BLOCK - 1) / WAVES_PER_BLOCK;
    if (blocks > 2048) blocks = 2048;                  // grid-stride; ~4-5 tiles/wave
    const int totalWaves = blocks * WAVES_PER_BLOCK;

    hipLaunchKernelGGL(edge_type_wmma_kernel, dim3(blocks),
                       dim3(WAVES_PER_BLOCK * 32), 0, stream,
                       X, EI, W, Bv, out, E, numTiles, totalWaves);
}